// InnovationLoss_88124138979690
// MI455X (gfx1250) — compile-verified
//
#include <hip/hip_runtime.h>
#include <math.h>

#define BB 8192
#define DD 2048
#define CCLS 1000
#define LL 256

typedef __attribute__((ext_vector_type(2))) float v2f;
typedef __attribute__((ext_vector_type(8))) float v8f;

// ---- workspace layout (float offsets) ----
#define WS_SUM_RGB   0                       // 256*2048
#define WS_SUM_SAR   (WS_SUM_RGB + LL*DD)    // 256*2048
#define WS_N_RGB     (WS_SUM_SAR + LL*DD)    // 256
#define WS_N_SAR     (WS_N_RGB + LL)         // 256
#define WS_ORTH      (WS_N_SAR + LL)         // 3
#define WS_P_RGB     (WS_ORTH + 3)           // 1000
#define WS_P_SAR     (WS_P_RGB + CCLS)       // 1000
#define WS_CNT       (WS_P_SAR + CCLS)       // 2
#define WS_MSEL      (WS_CNT + 2)            // 2: sum, valid count
#define WS_FLOATS    (WS_MSEL + 2)

__device__ __forceinline__ float block_reduce_sum(float v, float* s) {
  int t = threadIdx.x;
  s[t] = v;
  __syncthreads();
  #pragma unroll
  for (int off = 128; off > 0; off >>= 1) {
    if (t < off) s[t] += s[t + off];
    __syncthreads();
  }
  float r = s[0];
  __syncthreads();
  return r;
}

__device__ __forceinline__ float block_reduce_max(float v, float* s) {
  int t = threadIdx.x;
  s[t] = v;
  __syncthreads();
  #pragma unroll
  for (int off = 128; off > 0; off >>= 1) {
    if (t < off) s[t] = fmaxf(s[t], s[t + off]);
    __syncthreads();
  }
  float r = s[0];
  __syncthreads();
  return r;
}

__device__ __forceinline__ float dot4(float4 a, float4 b) {
  return a.x * b.x + a.y * b.y + a.z * b.z + a.w * b.w;
}

__global__ void zero_ws_kernel(float* ws, int n) {
  int i = blockIdx.x * blockDim.x + threadIdx.x;
  if (i < n) ws[i] = 0.f;
}

// Per-row L2 normalize feat_shared, scatter-add into per-(label,modal) sums.
__global__ void feat_scatter_kernel(const float* __restrict__ feat,
                                    const int* __restrict__ labels,
                                    const int* __restrict__ modal,
                                    float* ws) {
  __shared__ float sdata[256];
  int r = blockIdx.x, t = threadIdx.x;
  const float4* row = (const float4*)(feat + (size_t)r * DD);
  float4 x0 = row[t];
  float4 x1 = row[t + 256];
  float ss = dot4(x0, x0) + dot4(x1, x1);
  ss = block_reduce_sum(ss, sdata);
  float inv = 1.f / fmaxf(sqrtf(ss), 1e-12f);
  int lbl = labels[r];
  int m = modal[r];
  float* dst = ws + (m == 0 ? WS_SUM_RGB : WS_SUM_SAR) + (size_t)lbl * DD;
  int c0 = t * 4;
  atomicAdd(dst + c0 + 0, x0.x * inv);
  atomicAdd(dst + c0 + 1, x0.y * inv);
  atomicAdd(dst + c0 + 2, x0.z * inv);
  atomicAdd(dst + c0 + 3, x0.w * inv);
  atomicAdd(dst + c0 + 1024, x1.x * inv);
  atomicAdd(dst + c0 + 1025, x1.y * inv);
  atomicAdd(dst + c0 + 1026, x1.z * inv);
  atomicAdd(dst + c0 + 1027, x1.w * inv);
  if (t == 0) atomicAdd(ws + (m == 0 ? WS_N_RGB : WS_N_SAR) + lbl, 1.f);
}

// Per-row pairwise cosines of dee1/2/3; accumulate |cos| into 3 scalars.
__global__ void orth_kernel(const float* __restrict__ d1,
                            const float* __restrict__ d2,
                            const float* __restrict__ d3,
                            float* ws) {
  __shared__ float sdata[256];
  int r = blockIdx.x, t = threadIdx.x;
  const float4* a = (const float4*)(d1 + (size_t)r * DD);
  const float4* b = (const float4*)(d2 + (size_t)r * DD);
  const float4* c = (const float4*)(d3 + (size_t)r * DD);
  float4 a0 = a[t], a1 = a[t + 256];
  float4 b0 = b[t], b1 = b[t + 256];
  float4 c0 = c[t], c1 = c[t + 256];
  float s11 = dot4(a0, a0) + dot4(a1, a1);
  float s22 = dot4(b0, b0) + dot4(b1, b1);
  float s33 = dot4(c0, c0) + dot4(c1, c1);
  float s12 = dot4(a0, b0) + dot4(a1, b1);
  float s13 = dot4(a0, c0) + dot4(a1, c1);
  float s23 = dot4(b0, c0) + dot4(b1, c1);
  s11 = block_reduce_sum(s11, sdata);
  s22 = block_reduce_sum(s22, sdata);
  s33 = block_reduce_sum(s33, sdata);
  s12 = block_reduce_sum(s12, sdata);
  s13 = block_reduce_sum(s13, sdata);
  s23 = block_reduce_sum(s23, sdata);
  if (t == 0) {
    float den1 = fmaxf(sqrtf(s11), 1e-12f);
    float den2 = fmaxf(sqrtf(s22), 1e-12f);
    float den3 = fmaxf(sqrtf(s33), 1e-12f);
    atomicAdd(ws + WS_ORTH + 0, fabsf(s12) / (den1 * den2));
    atomicAdd(ws + WS_ORTH + 1, fabsf(s13) / (den1 * den3));
    atomicAdd(ws + WS_ORTH + 2, fabsf(s23) / (den2 * den3));
  }
}

// Softmax per row, accumulate per-modal mean probabilities. 64 blocks x 128 rows.
__global__ void consistency_kernel(const float* __restrict__ logits,
                                   const int* __restrict__ modal,
                                   float* ws) {
  __shared__ float acc[2][1024];
  __shared__ float sdata[256];
  __shared__ float scnt[2];
  int t = threadIdx.x;
  for (int cidx = t; cidx < 1024; cidx += 256) { acc[0][cidx] = 0.f; acc[1][cidx] = 0.f; }
  if (t < 2) scnt[t] = 0.f;
  __syncthreads();
  int rows = BB / gridDim.x;
  int r0 = blockIdx.x * rows;
  for (int r = r0; r < r0 + rows; ++r) {
    const float* lg = logits + (size_t)r * CCLS;
    if (r + 1 < r0 + rows)  // prefetch next row while reducing this one
      __builtin_prefetch(logits + (size_t)(r + 1) * CCLS + t * 4, 0, 1);
    float x[4];
    float mx = -3.4e38f;
    int ncol = 0;
    for (int cidx = t, j = 0; cidx < CCLS; cidx += 256, ++j) {
      x[j] = lg[cidx];
      mx = fmaxf(mx, x[j]);
      ncol = j + 1;
    }
    float rowmax = block_reduce_max(mx, sdata);
    float sume = 0.f;
    for (int j = 0; j < ncol; ++j) { x[j] = __expf(x[j] - rowmax); sume += x[j]; }
    float tot = block_reduce_sum(sume, sdata);
    float invt = 1.f / tot;
    int m = modal[r];
    for (int cidx = t, j = 0; cidx < CCLS; cidx += 256, ++j) acc[m][cidx] += x[j] * invt;
    if (t == 0) scnt[m] += 1.f;
  }
  __syncthreads();
  for (int cidx = t; cidx < CCLS; cidx += 256) {
    atomicAdd(ws + WS_P_RGB + cidx, acc[0][cidx]);
    atomicAdd(ws + WS_P_SAR + cidx, acc[1][cidx]);
  }
  if (t == 0) {
    atomicAdd(ws + WS_CNT + 0, scnt[0]);
    atomicAdd(ws + WS_CNT + 1, scnt[1]);
  }
}

// Diagonal 16x16 tiles of C_rgb*C_sar^T and C_rgb*C_rgb^T via V_WMMA_F32_16X16X4_F32.
// For a diagonal tile the documented A(16x4)/B(4x16) lane layouts give identical
// per-lane (row, k) addressing: lane loads float2 at row l0+(lane&15), col k0+(lane>>4)*2.
__global__ void msel_wmma_kernel(float* ws) {
  const float* sum_rgb = ws + WS_SUM_RGB;
  const float* sum_sar = ws + WS_SUM_SAR;
  const float* n_rgb = ws + WS_N_RGB;
  const float* n_sar = ws + WS_N_SAR;
  int lane = threadIdx.x;           // one wave (32 lanes) per block
  int l0 = blockIdx.x * 16;
  int row = l0 + (lane & 15);
  int kofs = (lane >> 4) << 1;
  const float* prg = sum_rgb + (size_t)row * DD + kofs;
  const float* psr = sum_sar + (size_t)row * DD + kofs;
  v8f accC = {};
  v8f accI = {};
  for (int k0 = 0; k0 < DD; k0 += 4) {
    v2f a = *(const v2f*)(prg + k0);
    v2f b = *(const v2f*)(psr + k0);
    accC = __builtin_amdgcn_wmma_f32_16x16x4_f32(false, a, false, b, (short)0, accC,
                                                 false, false);
    accI = __builtin_amdgcn_wmma_f32_16x16x4_f32(false, a, false, a, (short)0, accI,
                                                 false, false);
  }
  // diag element M=N=m: vgpr m%8 at lane m (m<8) or lane 24+(m-8) (m>=8)
  float gc = 0.f, gi = 0.f;
  int lbl = -1;
  #pragma unroll
  for (int i = 0; i < 8; ++i) {
    if (lane == i)      { gc = accC[i]; gi = accI[i]; lbl = l0 + i; }
    if (lane == 24 + i) { gc = accC[i]; gi = accI[i]; lbl = l0 + 8 + i; }
  }
  if (lbl >= 0) {
    float nr = n_rgb[lbl], ns = n_sar[lbl];
    float dr = fmaxf(nr, 1.f), dsn = fmaxf(ns, 1.f);
    float Gc = gc / (dr * dsn);          // c_rgb . c_sar
    float Gi = gi / (dr * dr);           // c_rgb . c_rgb
    bool valid = (nr >= 2.f) && (ns >= 1.f);
    float d = Gi - Gc;                   // dist_cross - dist_intra
    atomicAdd(ws + WS_MSEL + 0, valid ? d * d : 0.f);
    atomicAdd(ws + WS_MSEL + 1, valid ? 1.f : 0.f);
  }
}

__global__ void finalize_kernel(const float* __restrict__ ws, float* __restrict__ out) {
  __shared__ float sdata[256];
  int t = threadIdx.x;
  float nr = ws[WS_CNT + 0], ns = ws[WS_CNT + 1];
  float inr = 1.f / fmaxf(nr, 1.f);
  float ins = 1.f / fmaxf(ns, 1.f);
  float part = 0.f;
  for (int cidx = t; cidx < CCLS; cidx += 256) {
    float pr = ws[WS_P_RGB + cidx] * inr;
    float ps = ws[WS_P_SAR + cidx] * ins;
    float lr = __logf(pr), ls = __logf(ps);
    part += ps * (ls - lr) + pr * (lr - ls);
  }
  float kl = block_reduce_sum(part, sdata);
  if (t == 0) {
    float cons = (nr > 0.f && ns > 0.f) ? 0.5f * kl : 0.f;
    float orth = (ws[WS_ORTH + 0] + ws[WS_ORTH + 1] + ws[WS_ORTH + 2]) / (float)BB;
    float mc = ws[WS_MSEL + 1];
    float msel = (mc > 0.f) ? ws[WS_MSEL + 0] / fmaxf(mc, 1.f) : 0.f;
    out[0] = 0.5f * msel + 0.1f * orth + 0.1f * cons;
  }
}

extern "C" void kernel_launch(void* const* d_in, const int* in_sizes, int n_in,
                              void* d_out, int out_size, void* d_ws, size_t ws_size,
                              hipStream_t stream) {
  const float* feat   = (const float*)d_in[0];
  const float* dee1   = (const float*)d_in[1];
  const float* dee2   = (const float*)d_in[2];
  const float* dee3   = (const float*)d_in[3];
  const float* logits = (const float*)d_in[4];
  const int*   labels = (const int*)d_in[5];
  const int*   modal  = (const int*)d_in[6];
  float* ws  = (float*)d_ws;
  float* out = (float*)d_out;

  int nz = WS_FLOATS;
  zero_ws_kernel<<<(nz + 255) / 256, 256, 0, stream>>>(ws, nz);
  feat_scatter_kernel<<<BB, 256, 0, stream>>>(feat, labels, modal, ws);
  orth_kernel<<<BB, 256, 0, stream>>>(dee1, dee2, dee3, ws);
  consistency_kernel<<<64, 256, 0, stream>>>(logits, modal, ws);
  msel_wmma_kernel<<<16, 32, 0, stream>>>(ws);
  finalize_kernel<<<1, 256, 0, stream>>>(ws, out);
}